// model_1_180388626835
// MI455X (gfx1250) — compile-verified
//
#include <hip/hip_runtime.h>
#include <hip/hip_bf16.h>

// ---------------------------------------------------------------------------
// Hetero-GNN match predictor for MI455X (gfx1250, wave32).
//
// Algorithmic cut: reference returns (y, x_t_layer1) only -> player side of
// layer 1 (1M x 128 x 128 dual GEMM + 1M-row scatter) is dead and skipped.
// Pipeline is memory-bound (~2-3 GB vs 3.6e10 f32 FLOPs @ 23.3 TB/s), so we
// stay in f32: V_WMMA_F32_16X16X4_F32 for GEMMs, TDM (tensor_load_to_lds) for
// weight staging, global f32 atomics for the graph scatters.
// ---------------------------------------------------------------------------

typedef float v2f __attribute__((ext_vector_type(2)));
typedef float v8f __attribute__((ext_vector_type(8)));
typedef unsigned int v4u __attribute__((ext_vector_type(4)));
typedef int v4i __attribute__((ext_vector_type(4)));
typedef int v8i __attribute__((ext_vector_type(8)));

#define N_P   1000000
#define N_T   32768
#define NEDGE 1000000
#define D_EMB 64
#define HID   128
#define G_CNT 16384

// ---------------------------------------------------------------------------
// gather: out[row][:] = emb[ids[row]][:]   (one float4 per thread)
// ---------------------------------------------------------------------------
__global__ void gather_rows_kernel(const float* __restrict__ emb,
                                   const int* __restrict__ ids,
                                   float* __restrict__ out,
                                   int n, int logD) {
    long long t = (long long)blockIdx.x * blockDim.x + threadIdx.x;
    long long row = t >> (logD - 2);
    if (row >= n) return;
    int D = 1 << logD;
    int c = ((int)t & ((D >> 2) - 1)) << 2;
    const float4* s = (const float4*)(emb + (size_t)ids[row] * D + c);
    *(float4*)(out + (size_t)row * D + c) = *s;
}

__global__ void zero4_kernel(float4* __restrict__ p, long long n4) {
    long long t = (long long)blockIdx.x * blockDim.x + threadIdx.x;
    if (t < n4) p[t] = make_float4(0.f, 0.f, 0.f, 0.f);
}

// ---------------------------------------------------------------------------
// scatter-add: AGG[dst[e]][c] += X[src[e]][c]   (D = 64 or 128 channels)
// lanes cover consecutive channels of one edge -> coalesced row reads and
// contiguous global_atomic_add_f32 bursts (team aggregates stay L2-resident).
// ---------------------------------------------------------------------------
__global__ void scatter_add_kernel(const float* __restrict__ X,
                                   const int* __restrict__ src_idx,
                                   const int* __restrict__ dst_idx,
                                   float* __restrict__ AGG,
                                   int E, int logD) {
    long long t = (long long)blockIdx.x * blockDim.x + threadIdx.x;
    long long e = t >> logD;
    if (e >= E) return;
    int D = 1 << logD;
    int c = (int)t & (D - 1);
    int s = src_idx[e];
    int d = dst_idx[e];
    atomicAdd(&AGG[(size_t)d * D + c], X[(size_t)s * D + c]);
}

// ---------------------------------------------------------------------------
// TDM: DMA a 2D f32 tile (128 cols x nRows, row stride 128 elements) from
// global memory into LDS at byte offset ldsByteOff. Descriptor per ISA §8.3/8.4:
//   group0: count=1 | lds_addr | global_addr[56:0] | type=2
//   group1: data_size=4B, tensor_dim0=tile_dim0=128, tensor_dim1=tile_dim1=nRows,
//           tensor_dim0_stride=128, no multicast/pad/iterate
//   groups 2/3: zero (tile_dim2..4 = 0 -> unused)
// Issued once per wave (EXEC-independent); completion via s_wait_tensorcnt.
// ---------------------------------------------------------------------------
__device__ __forceinline__ void tdm_load_w128(const float* gptr,
                                              unsigned ldsByteOff, int nRows) {
    unsigned long long ga = (unsigned long long)(size_t)gptr;
    v4u g0;
    g0[0] = 1u;                                            // count=1, user D#
    g0[1] = ldsByteOff;                                    // lds_addr
    g0[2] = (unsigned)(ga & 0xffffffffu);                  // global_addr[31:0]
    g0[3] = (unsigned)((ga >> 32) & 0x1ffffffu) | (2u << 30); // [56:32] | type=2
    v8i g1;
    g1[0] = (int)(2u << 16);               // workgroup_mask=0, data_size=2 (4B)
    g1[1] = (int)(128u << 16);             // tensor_dim0[15:0]=128
    g1[2] = (int)((unsigned)nRows << 16);  // tensor_dim0[31:16]=0, tensor_dim1[15:0]=nRows
    g1[3] = (int)(128u << 16);             // tensor_dim1[31:16]=0, tile_dim0=128
    g1[4] = nRows;                         // tile_dim1=nRows, tile_dim2=0
    g1[5] = 128;                           // tensor_dim0_stride[31:0]=128
    g1[6] = 0;                             // stride0[47:32]=0, stride1[15:0]=0
    g1[7] = 0;
    v4i z4 = {0, 0, 0, 0};
#if defined(__clang_major__) && (__clang_major__ >= 23)
    v8i z8 = {0, 0, 0, 0, 0, 0, 0, 0};
    __builtin_amdgcn_tensor_load_to_lds(g0, g1, z4, z4, z8, 0);
#else
    __builtin_amdgcn_tensor_load_to_lds(g0, g1, z4, z4, 0);
#endif
}

// ---------------------------------------------------------------------------
// Fused dual GEMM + bias + ReLU:
//   Y[M,128] = relu( X0[M,K0] @ W0[K0,128] + X1[M,K1] @ W1[K1,128] + bias )
// i.e. concat([X0|X1]) @ concat_rows([W0;W1]) with Ktot = K0+K1 (128 or 256).
//
// 8 waves / 256-thread block; each wave owns a 16(M) x 128(N) tile with 8 v8f
// accumulators. W chunk (<=128 rows x 128 cols f32 = 64 KB) is TDM-DMA'd into
// LDS row-major by wave 0; B-fragments are ds_load_2addr_b32 pairs 128 DWORDs
// apart (consecutive lanes -> consecutive banks, conflict-free).
//
// WMMA f32 16x16x4 fragment layouts (ISA 7.12.2):
//   A: lanes 0-15 M=lane, {v0,v1}={K,K+1}; lanes 16-31 M=lane-16, {K+2,K+3}
//      -> contiguous float2 global load at column k + 2*(lane>>4)
//   B: mirrored per lane half; C/D: VGPR j -> (M=j / j+8, N=lane&15)
// ---------------------------------------------------------------------------
__global__ __launch_bounds__(256)
void gemm_dual_relu_kernel(const float* __restrict__ X0,
                           const float* __restrict__ X1,
                           const float* __restrict__ W0,
                           const float* __restrict__ W1,
                           const float* __restrict__ bias,
                           float* __restrict__ Y,
                           int M, int K0, int K1) {
    // Dynamic LDS, no static __shared__ in this kernel -> LDS byte offset 0,
    // which is what the TDM descriptors below assume.
    extern __shared__ float lds[];              // [Kc][128] row-major W chunk

    const int lane   = threadIdx.x & 31;
    const int wave   = threadIdx.x >> 5;
    const int tile   = blockIdx.x * 8 + wave;   // 16-row M tile
    const int mTiles = M >> 4;
    const bool active = (tile < mTiles);        // wave-uniform: EXEC stays all-1s
    const int rowBase = tile << 4;
    const int m    = lane & 15;
    const int half = lane >> 4;                 // selects K+2/K+3 sub-columns
    const int Ktot = K0 + K1;

    v8f acc[8] = {};

    for (int kc = 0; kc < Ktot; kc += 128) {
        const int Kc = (Ktot - kc) < 128 ? (Ktot - kc) : 128;
        const int r1 = kc + Kc;

        // ---- TDM-stage W chunk rows [kc, kc+Kc) of [W0;W1] into LDS -------
        if (wave == 0) {
            int nA = 0;
            if (kc < K0) {                       // rows from W0
                int aEnd = (r1 < K0) ? r1 : K0;
                nA = aEnd - kc;
                tdm_load_w128(W0 + (size_t)kc * HID, 0u, nA);
            }
            if (r1 > K0) {                       // rows from W1
                int s1 = ((kc > K0) ? kc : K0) - K0;
                int nB = r1 - ((kc > K0) ? kc : K0);
                tdm_load_w128(W1 + (size_t)s1 * HID,
                              (unsigned)nA * (HID * 4u), nB);
            }
            __builtin_amdgcn_s_wait_tensorcnt(0);
        }
        __syncthreads();

        if (active) {
            // stream-hint the agg operand for the upcoming chunk
            __builtin_prefetch(X1 + (size_t)(rowBase + m) * K1, 0, 0);

            for (int kk = 0; kk < Kc; kk += 4) {
                const int kg = kc + kk + 2 * half;      // never straddles K0 (K0 % 4 == 0)
                const float* asrc = (kg < K0)
                    ? (X0 + (size_t)(rowBase + m) * K0 + kg)
                    : (X1 + (size_t)(rowBase + m) * K1 + (kg - K0));
                v2f a;
                a[0] = asrc[0];
                a[1] = asrc[1];
#pragma unroll
                for (int nt = 0; nt < 8; ++nt) {
                    // row-major W chunk: W[k][n] at word k*128 + n
                    const float* bsrc = &lds[(kk + 2 * half) * HID + nt * 16 + m];
                    v2f b;
                    b[0] = bsrc[0];
                    b[1] = bsrc[HID];            // ds_load_2addr_b32 offset1:128
                    acc[nt] = __builtin_amdgcn_wmma_f32_16x16x4_f32(
                        false, a, false, b, (short)0, acc[nt], false, false);
                }
            }
        }
        __syncthreads();
    }

    if (active) {
#pragma unroll
        for (int nt = 0; nt < 8; ++nt) {
            const int col = nt * 16 + m;
            const float bb = bias[col];
#pragma unroll
            for (int j = 0; j < 8; ++j) {
                float v = acc[nt][j] + bb;
                v = v > 0.f ? v : 0.f;
                const int row = rowBase + j + half * 8;
                Y[(size_t)row * HID + col] = v;
            }
        }
    }
}

// ---------------------------------------------------------------------------
// head: y[g] = softmax( xt1[g].reshape(256) @ W_lin + b_lin )
// ---------------------------------------------------------------------------
__global__ void head_kernel(const float* __restrict__ xt1,   // [N_T, 128] = [G, 256]
                            const float* __restrict__ Wl,    // [256, 3]
                            const float* __restrict__ bl,    // [3]
                            float* __restrict__ y) {          // [G, 3]
    __shared__ float w[256 * 3];
    for (int i = threadIdx.x; i < 256 * 3; i += blockDim.x) w[i] = Wl[i];
    __syncthreads();

    int g = blockIdx.x * blockDim.x + threadIdx.x;
    if (g >= G_CNT) return;
    float a0 = bl[0], a1 = bl[1], a2 = bl[2];
    const float* x = xt1 + (size_t)g * 256;
#pragma unroll 4
    for (int k = 0; k < 256; ++k) {
        float v = x[k];
        a0 += v * w[k * 3 + 0];
        a1 += v * w[k * 3 + 1];
        a2 += v * w[k * 3 + 2];
    }
    float mx = fmaxf(a0, fmaxf(a1, a2));
    float e0 = expf(a0 - mx), e1 = expf(a1 - mx), e2 = expf(a2 - mx);
    float inv = 1.f / (e0 + e1 + e2);
    y[(size_t)g * 3 + 0] = e0 * inv;
    y[(size_t)g * 3 + 1] = e1 * inv;
    y[(size_t)g * 3 + 2] = e2 * inv;
}

// ---------------------------------------------------------------------------
// host orchestration
// ---------------------------------------------------------------------------
extern "C" void kernel_launch(void* const* d_in, const int* in_sizes, int n_in,
                              void* d_out, int out_size, void* d_ws, size_t ws_size,
                              hipStream_t stream) {
    const int*   player_ids = (const int*)  d_in[0];
    const int*   team_ids   = (const int*)  d_in[1];
    const int*   edge_src   = (const int*)  d_in[2];
    const int*   edge_dst   = (const int*)  d_in[3];
    const float* player_emb = (const float*)d_in[4];
    const float* team_emb   = (const float*)d_in[5];
    const float* W_rel_pt0  = (const float*)d_in[6];
    const float* b_rel_pt0  = (const float*)d_in[7];
    const float* W_root_pt0 = (const float*)d_in[8];
    const float* W_rel_tp0  = (const float*)d_in[9];
    const float* b_rel_tp0  = (const float*)d_in[10];
    const float* W_root_tp0 = (const float*)d_in[11];
    const float* W_rel_pt1  = (const float*)d_in[12];
    const float* b_rel_pt1  = (const float*)d_in[13];
    const float* W_root_pt1 = (const float*)d_in[14];
    // d_in[15..17] (W_rel_tp1 / b_rel_tp1 / W_root_tp1): dead — player layer 1 unused
    const float* W_lin      = (const float*)d_in[18];
    const float* b_lin      = (const float*)d_in[19];

    float* ws = (float*)d_ws;
    float* xp_emb = ws;                                  // [1M, 64]
    float* agg_p  = xp_emb + (size_t)N_P * D_EMB;        // [1M, 64]
    float* xp0    = agg_p  + (size_t)N_P * D_EMB;        // [1M, 128]
    float* xt_emb = xp0    + (size_t)N_P * HID;          // [32K, 64]
    float* agg_t0 = xt_emb + (size_t)N_T * D_EMB;        // [32K, 64]
    float* xt0    = agg_t0 + (size_t)N_T * D_EMB;        // [32K, 128]
    float* agg_t1 = xt0    + (size_t)N_T * HID;          // [32K, 128]

    float* y_out  = (float*)d_out;                       // [16384, 3]
    float* xt1    = y_out + (size_t)G_CNT * 3;           // [32K, 128] (second output)

    const int BLK = 256;
    const size_t gemm_lds = (size_t)128 * HID * sizeof(float);   // 64 KB

    // 1) embedding gathers
    gather_rows_kernel<<<(int)(((long long)N_P * 16 + BLK - 1) / BLK), BLK, 0, stream>>>(
        player_emb, player_ids, xp_emb, N_P, 6);
    gather_rows_kernel<<<(int)(((long long)N_T * 16 + BLK - 1) / BLK), BLK, 0, stream>>>(
        team_emb, team_ids, xt_emb, N_T, 6);

    // 2) clear aggregation buffers
    zero4_kernel<<<(int)(((long long)N_P * D_EMB / 4 + BLK - 1) / BLK), BLK, 0, stream>>>(
        (float4*)agg_p, (long long)N_P * D_EMB / 4);
    zero4_kernel<<<(int)(((long long)N_T * D_EMB / 4 + BLK - 1) / BLK), BLK, 0, stream>>>(
        (float4*)agg_t0, (long long)N_T * D_EMB / 4);
    zero4_kernel<<<(int)(((long long)N_T * HID / 4 + BLK - 1) / BLK), BLK, 0, stream>>>(
        (float4*)agg_t1, (long long)N_T * HID / 4);

    // 3) layer-0 aggregations
    scatter_add_kernel<<<(int)(((long long)NEDGE * 64 + BLK - 1) / BLK), BLK, 0, stream>>>(
        xp_emb, edge_src, edge_dst, agg_t0, NEDGE, 6);      // players -> teams
    scatter_add_kernel<<<(int)(((long long)NEDGE * 64 + BLK - 1) / BLK), BLK, 0, stream>>>(
        xt_emb, edge_dst, edge_src, agg_p, NEDGE, 6);       // teams -> players

    // 4) layer-0 dual GEMMs: new = relu(root@Wroot + agg@Wrel + b)
    {
        int mTiles = N_P >> 4;                               // 62500
        int grid = (mTiles + 7) / 8;                         // 7813
        gemm_dual_relu_kernel<<<grid, BLK, gemm_lds, stream>>>(
            xp_emb, agg_p, W_root_tp0, W_rel_tp0, b_rel_tp0, xp0, N_P, D_EMB, D_EMB);
    }
    {
        int grid = (N_T >> 4) / 8;                           // 256
        gemm_dual_relu_kernel<<<grid, BLK, gemm_lds, stream>>>(
            xt_emb, agg_t0, W_root_pt0, W_rel_pt0, b_rel_pt0, xt0, N_T, D_EMB, D_EMB);
    }

    // 5) layer-1 team aggregation (player layer-1 is dead code -> skipped)
    scatter_add_kernel<<<(int)(((long long)NEDGE * 128 + BLK - 1) / BLK), BLK, 0, stream>>>(
        xp0, edge_src, edge_dst, agg_t1, NEDGE, 7);

    // 6) layer-1 team dual GEMM -> writes x_t output directly
    {
        int grid = (N_T >> 4) / 8;                           // 256
        gemm_dual_relu_kernel<<<grid, BLK, gemm_lds, stream>>>(
            xt0, agg_t1, W_root_pt1, W_rel_pt1, b_rel_pt1, xt1, N_T, HID, HID);
    }

    // 7) classifier head + softmax
    head_kernel<<<(G_CNT + BLK - 1) / BLK, BLK, 0, stream>>>(xt1, W_lin, b_lin, y_out);
}